// HexGraphConv_79998060855868
// MI455X (gfx1250) — compile-verified
//
#include <hip/hip_runtime.h>

typedef __attribute__((ext_vector_type(2))) float v2f;
typedef __attribute__((ext_vector_type(4))) float v4f;
typedef __attribute__((ext_vector_type(8))) float v8f;

#define IN_DIM   128
#define OUT_DIM  128
#define KTOT     256           // concat K: [x | S/denom]
#define NEG_SLOPE 0.1f

#define WC_STRIDE 132          // pad 128->132: rows k,k+2 hit different banks
#define A_STRIDE  260          // pad 256->260: 16 rows fan across banks

// ---------------------------------------------------------------------------
// Phase 1: edge scatter.  S[b,dst,:] += x[b,src,:]  and indeg[dst] += 1.
// One thread per (edge, channel); loop over batch. Coalesced 128B gathers per
// wave; fp32 global atomics land in L2 (192MB -> whole S buffer is resident).
// ---------------------------------------------------------------------------
__global__ __launch_bounds__(256) void
hexconv_scatter(const float* __restrict__ x,
                const long long* __restrict__ ei,   // (2,E) int64
                float* __restrict__ S,              // (B*N,128) pre-zeroed
                int* __restrict__ indeg,            // (N) pre-zeroed
                int E, int N, int B)
{
    long long tid = (long long)blockIdx.x * blockDim.x + threadIdx.x;
    long long e = tid >> 7;
    int c = (int)(tid & 127);
    if (e >= E) return;

    int src = (int)ei[e];
    int dst = (int)ei[(long long)E + e];
    if (c == 0) atomicAdd(&indeg[dst], 1);

    #pragma unroll 4
    for (int b = 0; b < B; ++b) {
        float v = x[((long long)b * N + src) * IN_DIM + c];
        atomicAdd(&S[((long long)b * N + dst) * IN_DIM + c], v);
    }
}

// ---------------------------------------------------------------------------
// Phase 2: persistent fused GEMM + bias + LeakyReLU, in place over S (=d_out).
//   out_row = leaky( [x_row | S_row/denom] @ [[W_self],[W_neigh]]
//                    + b_self + (indeg/denom) * b_neigh )
// Weights staged into LDS ONCE per block; block grid-strides over 16-row
// tiles. 8 waves per block, each owns a 16-col tile and chains 64x
// V_WMMA_F32_16X16X4_F32 per tile.
// ---------------------------------------------------------------------------
__global__ __launch_bounds__(256) void
hexconv_gemm(const float* __restrict__ x,
             float* __restrict__ out,               // in: S sums, out: result
             const long long* __restrict__ deg,     // (N) int64
             const int* __restrict__ indeg,         // (N)
             const float* __restrict__ W_self,      // (128,128) row=c, col=o
             const float* __restrict__ b_self,      // (128)
             const float* __restrict__ W_neigh,     // (128,128)
             const float* __restrict__ b_neigh,     // (128)
             int N, int M, int numTiles)            // M = B*N
{
    __shared__ float wc[KTOT * WC_STRIDE];          // 135168 B
    __shared__ float alds[16 * A_STRIDE];           //  16640 B
    __shared__ float invd_s[16];
    __shared__ float scale_s[16];

    const int tid = threadIdx.x;

    // Stage combined weight once: rows 0..127 = W_self, 128..255 = W_neigh.
    for (int i = tid; i < KTOT * OUT_DIM / 4; i += 256) {
        int k = (i * 4) / OUT_DIM;
        int o = (i * 4) % OUT_DIM;
        const v4f v = (k < IN_DIM)
            ? *(const v4f*)(W_self  + (size_t)k * OUT_DIM + o)
            : *(const v4f*)(W_neigh + (size_t)(k - IN_DIM) * OUT_DIM + o);
        float* dp = &wc[k * WC_STRIDE + o];
        dp[0] = v.x; dp[1] = v.y; dp[2] = v.z; dp[3] = v.w;
    }

    const int lane    = tid & 31;
    const int wave    = tid >> 5;      // 0..7 -> column tile
    const int lhalf   = lane >> 4;     // 0: K={k0,k0+1}, 1: K={k0+2,k0+3}
    const int l15     = lane & 15;
    const int colbase = wave * 16;
    const int col     = colbase + l15;
    const float bs    = b_self[col];
    const float bn    = b_neigh[col];

    for (int t = blockIdx.x; t < numTiles; t += gridDim.x) {
        const int rowbase = t * 16;

        __syncthreads();   // previous tile's consumers of alds/scale_s done

        // Per-row parameters (denom = clip(deg,1); scale = indeg/denom).
        if (tid < 16) {
            int r = rowbase + tid; if (r >= M) r = M - 1;
            int n = r % N;
            float d = (float)deg[n];
            d = d < 1.0f ? 1.0f : d;
            float inv = 1.0f / d;
            invd_s[tid]  = inv;
            scale_s[tid] = (float)indeg[n] * inv;
        }

        __syncthreads();   // invd_s visible (and wc on first trip)

        // Stage A panel: 16 rows x 256 (x half raw, S half scaled by 1/denom).
        // Streamed once -> non-temporal loads keep caches for weights/atomics.
        for (int i = tid; i < 16 * KTOT / 4; i += 256) {
            int row = (i * 4) / KTOT;
            int k   = (i * 4) % KTOT;
            int r = rowbase + row; if (r >= M) r = M - 1;
            v4f v;
            if (k < IN_DIM) {
                v = __builtin_nontemporal_load(
                        (const v4f*)(x + (long long)r * IN_DIM + k));
            } else {
                v = __builtin_nontemporal_load(
                        (const v4f*)(out + (long long)r * IN_DIM + (k - IN_DIM)));
                float s = invd_s[row];
                v.x *= s; v.y *= s; v.z *= s; v.w *= s;
            }
            float* dp = &alds[row * A_STRIDE + k];
            dp[0] = v.x; dp[1] = v.y; dp[2] = v.z; dp[3] = v.w;
        }

        __syncthreads();   // A panel visible

        v8f acc = {};
        const float* arow = &alds[l15 * A_STRIDE];

        #pragma unroll 8
        for (int k0 = 0; k0 < KTOT; k0 += 4) {
            int ka = k0 + (lhalf ? 2 : 0);
            v2f a, b;
            a.x = arow[ka];
            a.y = arow[ka + 1];
            b.x = wc[ka       * WC_STRIDE + col];
            b.y = wc[(ka + 1) * WC_STRIDE + col];
            acc = __builtin_amdgcn_wmma_f32_16x16x4_f32(
                      /*neg_a=*/false, a, /*neg_b=*/false, b,
                      /*c_mod=*/(short)0, acc,
                      /*reuse_a=*/false, /*reuse_b=*/false);
        }

        // Epilogue: bias + LeakyReLU. C/D layout: VGPR j -> M = j + 8*lhalf.
        float* po = out + ((long long)(rowbase + 8 * lhalf)) * OUT_DIM + col;
        if (rowbase + 16 <= M) {
            // Fast path: unconditional stores at constant immediate offsets.
            #pragma unroll
            for (int j = 0; j < 8; ++j) {
                int rl = j + (lhalf ? 8 : 0);
                float v = acc[j] + bs + scale_s[rl] * bn;
                v = (v >= 0.0f) ? v : NEG_SLOPE * v;
                __builtin_nontemporal_store(v, po + j * OUT_DIM);
            }
        } else {
            #pragma unroll
            for (int j = 0; j < 8; ++j) {
                int rl = j + (lhalf ? 8 : 0);
                if (rowbase + rl < M) {
                    float v = acc[j] + bs + scale_s[rl] * bn;
                    v = (v >= 0.0f) ? v : NEG_SLOPE * v;
                    __builtin_nontemporal_store(v, po + j * OUT_DIM);
                }
            }
        }
    }
}

// ---------------------------------------------------------------------------
extern "C" void kernel_launch(void* const* d_in, const int* in_sizes, int n_in,
                              void* d_out, int out_size, void* d_ws, size_t ws_size,
                              hipStream_t stream)
{
    const float*     x       = (const float*)d_in[0];
    const long long* ei      = (const long long*)d_in[1];
    const long long* deg     = (const long long*)d_in[2];
    const float*     W_self  = (const float*)d_in[3];
    const float*     b_self  = (const float*)d_in[4];
    const float*     W_neigh = (const float*)d_in[5];
    const float*     b_neigh = (const float*)d_in[6];
    float*           out     = (float*)d_out;

    const int N = in_sizes[2];
    const int E = in_sizes[1] / 2;
    const int B = in_sizes[0] / (N * IN_DIM);
    const int M = B * N;

    int* indeg = (int*)d_ws;

    // Zero the S accumulator (aliased with d_out) and indeg each call.
    hipMemsetAsync(d_out, 0, (size_t)out_size * sizeof(float), stream);
    hipMemsetAsync(indeg, 0, (size_t)N * sizeof(int), stream);

    // Phase 1: scatter.
    long long total = (long long)E * IN_DIM;
    int sblocks = (int)((total + 255) / 256);
    hexconv_scatter<<<sblocks, 256, 0, stream>>>(x, ei, out, indeg, E, N, B);

    // Phase 2: persistent fused WMMA GEMM + epilogue.
    int numTiles = (M + 15) / 16;
    int gblocks = numTiles < 1024 ? numTiles : 1024;
    hexconv_gemm<<<gblocks, 256, 0, stream>>>(x, out, deg, indeg,
                                              W_self, b_self, W_neigh, b_neigh,
                                              N, M, numTiles);
}